// NonlinearWHT_44255343018500
// MI455X (gfx1250) — compile-verified
//
#include <hip/hip_runtime.h>
#include <hip/hip_bf16.h>
#include <stdint.h>

// ---------------------------------------------------------------------------
// Nonlinear Walsh-Hadamard transform, MI455X (gfx1250).
//
// 256 threads/block, one 8192-float row per block. TDM DMAs the row into LDS
// with a 4-dword skew every 32 dwords (pad_enable) so all LDS access phases
// are bank-conflict-free(-minimal). Each thread keeps 32 elements in VGPRs:
//   phase 1: stages 0-4  (index bits 0-4 free per thread)
//   phase 2: stages 5-8  (bits 5-8 free, bit 9 rides)
//   phase 3: stages 9-12 (bits 9-12 free, bit 0 rides)
// with only two LDS reorg shuffles between phases (6 LDS dwords/element vs 26
// for the naive in-LDS version, 4 barriers vs 13). Final result is written
// contiguously and DMAed back with tensor_store_from_lds.
// Bottleneck: ~436M v_tanh_f32 on the TRANS pipe; HBM floor 256MB ~ 11us.
// ---------------------------------------------------------------------------

#define WHT_N        8192
#define WHT_THREADS  256
#define WHT_LDSDW    9216            // 8192 data dwords + 256*4 skew dwords

typedef __attribute__((ext_vector_type(4))) unsigned int u32x4;
typedef __attribute__((ext_vector_type(8))) int          i32x8;
typedef __attribute__((ext_vector_type(4))) int          i32x4;

__device__ __forceinline__ float fast_tanh(float x) {
#if __has_builtin(__builtin_amdgcn_tanhf)
    return __builtin_amdgcn_tanhf(x);
#else
    float r;
    asm volatile("v_tanh_f32 %0, %1" : "=v"(r) : "v"(x));
    return r;
#endif
}

// TDM descriptor groups 0/1 for a contiguous 1-row (n x 1) fp32 tile.
// Layout per CDNA5 ISA sec. 8.3/8.4. 'skew' enables LDS padding of 4 dwords
// every 32 dwords (pad_interval code 4, pad_amount code 3) -- load only.
__device__ __forceinline__ void tdm_row_desc(u32x4& g0, i32x8& g1,
                                             unsigned lds_addr,
                                             unsigned long long gaddr,
                                             unsigned n_elems, bool skew) {
    g0.x = 1u;                                     // count = 1 valid descriptor
    g0.y = lds_addr;                               // lds_addr (bytes)
    g0.z = (unsigned)(gaddr & 0xFFFFFFFFull);      // global_addr[31:0]
    g0.w = (unsigned)((gaddr >> 32) & 0x01FFFFFFu) // global_addr[56:32]
         | (2u << 30);                             // type = 2 ("image")

    unsigned w0 = (2u << 16);                      // data_size = 2 -> 4 bytes
    if (skew)
        w0 |= (1u << 20)                           // pad_enable
            | (4u << 22)                           // pad_interval: 32 dwords
            | (3u << 25);                          // pad_amount:   4 dwords
    g1.s0 = (int)w0;
    g1.s1 = (int)((n_elems & 0xFFFFu) << 16);      // tensor_dim0[15:0] @63:48
    g1.s2 = (int)(((n_elems >> 16) & 0xFFFFu)      // tensor_dim0[31:16] @79:64
                  | (1u << 16));                   // tensor_dim1 = 1 @95:80
    g1.s3 = (int)((n_elems & 0xFFFFu) << 16);      // tile_dim0 = n @127:112
    g1.s4 = 1;                                     // tile_dim1 = 1 @143:128
    g1.s5 = (int)n_elems;                          // tensor_dim0_stride[31:0]
    g1.s6 = 0;
    g1.s7 = 0;
}

__device__ __forceinline__ void tdm_load(const u32x4& g0, const i32x8& g1) {
    i32x4 z4 = {0, 0, 0, 0};
#if defined(__clang_major__) && (__clang_major__ >= 23)
    i32x8 z8 = {0, 0, 0, 0, 0, 0, 0, 0};
    __builtin_amdgcn_tensor_load_to_lds(g0, g1, z4, z4, z8, 0);
#else
    __builtin_amdgcn_tensor_load_to_lds(g0, g1, z4, z4, 0);
#endif
}

__device__ __forceinline__ void tdm_store(const u32x4& g0, const i32x8& g1) {
    i32x4 z4 = {0, 0, 0, 0};
#if defined(__clang_major__) && (__clang_major__ >= 23)
    i32x8 z8 = {0, 0, 0, 0, 0, 0, 0, 0};
    __builtin_amdgcn_tensor_store_from_lds(g0, g1, z4, z4, z8, 0);
#else
    __builtin_amdgcn_tensor_store_from_lds(g0, g1, z4, z4, 0);
#endif
}

__global__ __launch_bounds__(WHT_THREADS) void
nonlinear_wht_kernel(const float* __restrict__ x, float* __restrict__ y) {
    __shared__ float lds[WHT_LDSDW];              // 36 KB of 320 KB WGP LDS

    const int      t       = (int)threadIdx.x;    // 0..255
    const unsigned wave_id = (unsigned)t >> 5;    // wave32
    const unsigned lds_off = (unsigned)(uintptr_t)(&lds[0]); // flat[31:0] == LDS byte offset

    const unsigned long long row_bytes = (unsigned long long)WHT_N * 4ull;
    const unsigned long long src =
        (unsigned long long)(uintptr_t)x + (unsigned long long)blockIdx.x * row_bytes;
    const unsigned long long dst =
        (unsigned long long)(uintptr_t)y + (unsigned long long)blockIdx.x * row_bytes;

    // ---- async DMA: global row -> LDS with 36-dword skewed chunks ----
    if (wave_id == 0) {
        u32x4 g0; i32x8 g1;
        tdm_row_desc(g0, g1, lds_off, src, WHT_N, /*skew=*/true);
        tdm_load(g0, g1);
        __builtin_amdgcn_s_wait_tensorcnt(0);
    }
    __syncthreads();

    const float inv_s2 = 0.70710678118654752440f;  // 1/sqrt(2)
    const float gain   = 1.6666666666666667f;      // 5/3

    float v[32];

#define BFLY(A, B)                                          \
    {                                                       \
        const float a_ = v[(A)], b_ = v[(B)];               \
        v[(A)] = gain * fast_tanh((a_ + b_) * inv_s2);      \
        v[(B)] = gain * fast_tanh((a_ - b_) * inv_s2);      \
    }

    // ================= Phase 1: stages 0-4 (e = t*32 + j) =================
    // skewed addr(e) = e + 4*(e>>5) = 36*t + j  -> b128 loads, 16B aligned
    {
        const int base1 = 36 * t;
#pragma unroll
        for (int q = 0; q < 8; ++q) {
            const float4 f = *(const float4*)&lds[base1 + 4 * q];
            v[4 * q + 0] = f.x; v[4 * q + 1] = f.y;
            v[4 * q + 2] = f.z; v[4 * q + 3] = f.w;
        }
#pragma unroll
        for (int s = 0; s < 5; ++s) {
            const int hb = 1 << s;
#pragma unroll
            for (int j = 0; j < 32; ++j)
                if ((j & hb) == 0) BFLY(j, j | hb)
        }
        // reorg1 write-back to own (phase-1) addresses; no barrier needed
        // between own-read and own-write.
#pragma unroll
        for (int q = 0; q < 8; ++q) {
            const float4 f = {v[4 * q + 0], v[4 * q + 1],
                              v[4 * q + 2], v[4 * q + 3]};
            *(float4*)&lds[base1 + 4 * q] = f;
        }
    }
    __syncthreads();

    // ====== Phase 2: stages 5-8 (e = (t>>5)<<10 | j<<5 | (t&31)) ==========
    // skewed addr = (t>>5)*1152 + 36*j + (t&31)  -> lane-consecutive b32
    {
        const int base2 = (t >> 5) * 1152 + (t & 31);
#pragma unroll
        for (int j = 0; j < 32; ++j)
            v[j] = lds[base2 + 36 * j];
#pragma unroll
        for (int s = 0; s < 4; ++s) {              // e-bits 5..8 == j-bits 0..3
            const int hb = 1 << s;
#pragma unroll
            for (int j = 0; j < 32; ++j)
                if ((j & hb) == 0) BFLY(j, j | hb)
        }
#pragma unroll
        for (int j = 0; j < 32; ++j)
            lds[base2 + 36 * j] = v[j];
    }
    __syncthreads();

    // ====== Phase 3: stages 9-12 (e = (j>>1)<<9 | t<<1 | (j&1)) ===========
    // skewed addr = (j>>1)*576 + 2*t + 4*(t>>4) + (j&1) -> stride-2 b64
    {
        const int base3 = 2 * t + 4 * (t >> 4);
#pragma unroll
        for (int m = 0; m < 16; ++m) {
            const float2 f = *(const float2*)&lds[m * 576 + base3];
            v[2 * m + 0] = f.x;
            v[2 * m + 1] = f.y;
        }
#pragma unroll
        for (int s = 0; s < 4; ++s) {              // e-bits 9..12 == j-bits 1..4
            const int hb = 2 << s;
#pragma unroll
            for (int j = 0; j < 32; ++j)
                if ((j & hb) == 0) BFLY(j, j | hb)
        }
        // final write: CONTIGUOUS (un-skewed) layout for the TDM store
        // addr(e) = e = m*512 + 2*t (+1); stride-2 b64, conflict-free
#pragma unroll
        for (int m = 0; m < 16; ++m) {
            const float2 f = {v[2 * m + 0], v[2 * m + 1]};
            *(float2*)&lds[m * 512 + 2 * t] = f;
        }
    }
    __syncthreads();
#undef BFLY

    // ---- async DMA: LDS (contiguous) -> global row ----
    if (wave_id == 0) {
        u32x4 g0; i32x8 g1;
        tdm_row_desc(g0, g1, lds_off, dst, WHT_N, /*skew=*/false);
        tdm_store(g0, g1);
        __builtin_amdgcn_s_wait_tensorcnt(0);
    }
}

extern "C" void kernel_launch(void* const* d_in, const int* in_sizes, int n_in,
                              void* d_out, int out_size, void* d_ws, size_t ws_size,
                              hipStream_t stream) {
    (void)n_in; (void)out_size; (void)d_ws; (void)ws_size;
    const float* x = (const float*)d_in[0];
    float*       y = (float*)d_out;
    const int rows = in_sizes[0] / WHT_N;         // 4096 for the reference shape
    if (rows <= 0) return;
    nonlinear_wht_kernel<<<dim3((unsigned)rows), dim3(WHT_THREADS), 0, stream>>>(x, y);
}